// RetinaNetPostProcessor_49048526520736
// MI455X (gfx1250) — compile-verified
//
#include <hip/hip_runtime.h>
#include <stdint.h>

typedef __attribute__((ext_vector_type(2))) float v2f;
typedef __attribute__((ext_vector_type(8))) float v8f;

namespace {
constexpr int NIMG = 2, AA = 9, CC = 80, HH = 100, WW_ = 152;
constexpr int HW   = HH * WW_;          // 15200
constexpr int CHW  = CC * HW;           // 1216000
constexpr int PER_IMG  = AA * CC * HW;  // 10944000
constexpr int PER_IMG4 = PER_IMG / 4;   // 2736000
constexpr int TOT4 = NIMG * PER_IMG4;   // 5472000
constexpr int KTOP = 1000, KPAD = 1024;
constexpr float LOGIT_TH = -2.94443897916644046f;  // ln(0.05/0.95)
constexpr float XCLIP    =  4.13516655674235609f;  // ln(1000/16)
constexpr float IMW_M1 = 1215.0f, IMH_M1 = 799.0f;
constexpr float CLS_OFF = 4096.0f;

// workspace offsets, in u32 units (total ~1.37 MB)
constexpr size_t HIST1 = 0;            // 2*65536
constexpr size_t HIST2 = 131072;       // 2*65536
constexpr size_t SEL   = 262144;       // 2*8  [b1,cnt1,mode,Tkey,cntGt,takeEq,Ksel,pad]
constexpr size_t CNT   = 262160;       // 2*2
constexpr size_t ZERO_W= 262208;       // zero everything below here each call
constexpr size_t CANDV = 262208;       // 2*1024 float
constexpr size_t CANDI = 264256;       // 2*1024 u32
constexpr size_t SORTV = 266304;       // 2*1024 float
constexpr size_t SORTI = 268352;       // 2*1024 u32
constexpr size_t BOXES = 270400;       // 2*1024*4 float (16B aligned)
constexpr size_t BOXOFF= 278592;       // 2*1024*4 float
constexpr size_t SCORES= 286784;       // 2*1024 float
constexpr size_t LABELS= 288832;       // 2*1024 u32
constexpr size_t VALIDF= 290880;       // 2*1024 u32
constexpr size_t MASKW = 292928;       // 2*1024*32 u32 suppression bitmask
} // namespace

__device__ __forceinline__ unsigned ordKey(float x) {
  unsigned u = __float_as_uint(x);
  return u ^ ((u & 0x80000000u) ? 0xFFFFFFFFu : 0x80000000u);
}

// ---------------- K0: zero histograms / counters ----------------
__global__ void k0_zero(unsigned* __restrict__ ws) {
  for (size_t i = (size_t)blockIdx.x * blockDim.x + threadIdx.x; i < ZERO_W;
       i += (size_t)gridDim.x * blockDim.x)
    ws[i] = 0u;
}

// ---------------- K1: histogram of high-16 key bits ----------------
__global__ void k1_hist(const float4* __restrict__ cls4, unsigned* __restrict__ ws) {
  unsigned* hist = ws + HIST1;
  for (int i = blockIdx.x * blockDim.x + threadIdx.x; i < TOT4;
       i += gridDim.x * blockDim.x) {
    int n = i / PER_IMG4;
    float4 v = cls4[i];
    float vv[4] = {v.x, v.y, v.z, v.w};
#pragma unroll
    for (int j = 0; j < 4; ++j) {
      float x = vv[j];
      if (x > LOGIT_TH)
        atomicAdd(&hist[(size_t)n * 65536 + (ordKey(x) >> 16)], 1u);
    }
  }
}

// ---------------- K2: select cut bin from hist1 ----------------
__global__ __launch_bounds__(1024) void k2_sel1(unsigned* __restrict__ ws) {
  int img = blockIdx.x, tid = threadIdx.x;
  const unsigned* h = ws + HIST1 + (size_t)img * 65536;
  __shared__ unsigned part[1024];
  unsigned s = 0;
#pragma unroll 4
  for (int k = 0; k < 64; ++k) s += h[tid * 64 + k];
  part[tid] = s;
  __syncthreads();
  if (tid == 0) {
    unsigned* sel = ws + SEL + (size_t)img * 8;
    unsigned cum = 0; int seg = -1;
    for (int t = 1023; t >= 0; --t) {
      if (cum + part[t] >= (unsigned)KTOP) { seg = t; break; }
      cum += part[t];
    }
    if (seg < 0) { // fewer than K candidates total -> take all
      sel[0] = 0xFFFFFFFFu; sel[1] = cum; sel[2] = 1u;
    } else {
      unsigned c2 = cum; int b = -1;
      for (int bin = seg * 64 + 63; bin >= seg * 64; --bin) {
        if (c2 + h[bin] >= (unsigned)KTOP) { b = bin; break; }
        c2 += h[bin];
      }
      sel[0] = (unsigned)b; sel[1] = c2; sel[2] = 0u;
    }
  }
}

// ---------------- K3: histogram of low-16 bits within cut bin ----------------
__global__ void k3_hist2(const float4* __restrict__ cls4, unsigned* __restrict__ ws) {
  unsigned* hist2 = ws + HIST2;
  for (int i = blockIdx.x * blockDim.x + threadIdx.x; i < TOT4;
       i += gridDim.x * blockDim.x) {
    int n = i / PER_IMG4;
    unsigned b1 = ws[SEL + (size_t)n * 8 + 0];
    float4 v = cls4[i];
    float vv[4] = {v.x, v.y, v.z, v.w};
#pragma unroll
    for (int j = 0; j < 4; ++j) {
      float x = vv[j];
      if (x > LOGIT_TH) {
        unsigned k = ordKey(x);
        if ((k >> 16) == b1)
          atomicAdd(&hist2[(size_t)n * 65536 + (k & 0xFFFFu)], 1u);
      }
    }
  }
}

// ---------------- K4: exact 32-bit threshold key ----------------
__global__ __launch_bounds__(1024) void k4_sel2(unsigned* __restrict__ ws) {
  int img = blockIdx.x, tid = threadIdx.x;
  unsigned* sel = ws + SEL + (size_t)img * 8;
  const unsigned* h = ws + HIST2 + (size_t)img * 65536;
  __shared__ unsigned part[1024];
  unsigned s = 0;
#pragma unroll 4
  for (int k = 0; k < 64; ++k) s += h[tid * 64 + k];
  part[tid] = s;
  __syncthreads();
  if (tid == 0) {
    if (sel[2]) { // take-all mode
      sel[3] = ordKey(LOGIT_TH);
      sel[4] = sel[1]; sel[5] = 0u;
      sel[6] = sel[1] < (unsigned)KTOP ? sel[1] : (unsigned)KTOP;
    } else {
      unsigned cum = sel[1]; int seg = -1;
      for (int t = 1023; t >= 0; --t) {
        if (cum + part[t] >= (unsigned)KTOP) { seg = t; break; }
        cum += part[t];
      }
      int b = 0;
      for (int bin = seg * 64 + 63; bin >= seg * 64; --bin) {
        if (cum + h[bin] >= (unsigned)KTOP) { b = bin; break; }
        cum += h[bin];
      }
      sel[3] = (sel[0] << 16) | (unsigned)b;
      sel[4] = cum;                    // count strictly greater than Tkey
      sel[5] = (unsigned)KTOP - cum;   // take from equal set
      sel[6] = (unsigned)KTOP;
    }
  }
}

// ---------------- K5: compaction ----------------
__global__ void k5_compact(const float4* __restrict__ cls4, unsigned* __restrict__ ws) {
  float* candV = (float*)(ws + CANDV);
  unsigned* candI = ws + CANDI;
  unsigned* cnt = ws + CNT;
  for (int i = blockIdx.x * blockDim.x + threadIdx.x; i < TOT4;
       i += gridDim.x * blockDim.x) {
    int n = i / PER_IMG4;
    int local4 = i - n * PER_IMG4;
    unsigned Tk = ws[SEL + (size_t)n * 8 + 3];
    unsigned cg = ws[SEL + (size_t)n * 8 + 4];
    unsigned te = ws[SEL + (size_t)n * 8 + 5];
    float4 v = cls4[i];
    float vv[4] = {v.x, v.y, v.z, v.w};
#pragma unroll
    for (int j = 0; j < 4; ++j) {
      float x = vv[j];
      if (x > LOGIT_TH) {
        unsigned k = ordKey(x);
        int slot = -1;
        if (k > Tk) {
          slot = (int)atomicAdd(&cnt[n * 2 + 0], 1u);
        } else if (k == Tk) {
          unsigned q = atomicAdd(&cnt[n * 2 + 1], 1u);
          if (q < te) slot = (int)(cg + q);
        }
        if (slot >= 0 && slot < KPAD) {
          int im = local4 * 4 + j;                  // (a,c,h,w) flat within image
          int a = im / CHW; int rem = im - a * CHW;
          int c = rem / HW; int hw = rem - c * HW;
          unsigned idxRef = ((unsigned)(hw * AA + a)) * CC + (unsigned)c;
          candV[(size_t)n * KPAD + slot] = x;
          candI[(size_t)n * KPAD + slot] = idxRef;
        }
      }
    }
  }
}

// ---------------- shared bitonic sort (descending score, ties by idx asc) ----------------
__device__ __forceinline__ void bitonicSortDesc1024(float* key, int* idx, int tid) {
  for (int k = 2; k <= 1024; k <<= 1) {
    for (int j = k >> 1; j > 0; j >>= 1) {
      __syncthreads();
      int ixj = tid ^ j;
      if (ixj > tid) {
        float ka = key[tid], kb = key[ixj];
        int ia = idx[tid], ib = idx[ixj];
        bool aBeforeB = (ka > kb) || (ka == kb && ia < ib);
        bool up = (tid & k) != 0;                 // ascending sub-block
        bool doSwap = up ? aBeforeB : !aBeforeB;  // overall: descending
        if (doSwap) { key[tid] = kb; key[ixj] = ka; idx[tid] = ib; idx[ixj] = ia; }
      }
    }
  }
  __syncthreads();
}

// ---------------- K6a: sort candidates ----------------
__global__ __launch_bounds__(1024) void k6a_sort(unsigned* __restrict__ ws) {
  int img = blockIdx.x, tid = threadIdx.x;
  __shared__ float sk[1024];
  __shared__ int   si[1024];
  unsigned ksel = ws[SEL + (size_t)img * 8 + 6];
  const float* candV = (const float*)(ws + CANDV) + (size_t)img * KPAD;
  const unsigned* candI = ws + CANDI + (size_t)img * KPAD;
  if ((unsigned)tid < ksel) { sk[tid] = candV[tid]; si[tid] = (int)candI[tid]; }
  else                      { sk[tid] = -3.0e38f;   si[tid] = 0x7FFFFFFF; }
  bitonicSortDesc1024(sk, si, tid);
  ((float*)(ws + SORTV))[(size_t)img * KPAD + tid] = sk[tid];
  ws[SORTI + (size_t)img * KPAD + tid] = (unsigned)si[tid];
}

// ---------------- K6: decode boxes ----------------
__global__ __launch_bounds__(1024) void k6_decode(const float* __restrict__ reg,
                                                  const float* __restrict__ anc,
                                                  unsigned* __restrict__ ws) {
  int img = blockIdx.x, s = threadIdx.x;
  unsigned ksel = ws[SEL + (size_t)img * 8 + 6];
  float* boxes = (float*)(ws + BOXES)  + ((size_t)img * KPAD + s) * 4;
  float* boff  = (float*)(ws + BOXOFF) + ((size_t)img * KPAD + s) * 4;
  float* scores = (float*)(ws + SCORES) + (size_t)img * KPAD;
  unsigned* labels = ws + LABELS + (size_t)img * KPAD;
  unsigned* valid  = ws + VALIDF + (size_t)img * KPAD;
  if ((unsigned)s < ksel) {
    float logit = ((const float*)(ws + SORTV))[(size_t)img * KPAD + s];
    unsigned idx = ws[SORTI + (size_t)img * KPAD + s];
    int loc = (int)(idx / CC); int cls = (int)idx - loc * CC; int label = cls + 1;
    int hw = loc / AA; int a = loc - hw * AA;
    const float* ab = anc + (size_t)loc * 4;
    float ax1 = ab[0], ay1 = ab[1], ax2 = ab[2], ay2 = ab[3];
    const float* rb = reg + (size_t)img * AA * 4 * HW + hw;
    float r0 = rb[(a * 4 + 0) * HW], r1 = rb[(a * 4 + 1) * HW];
    float r2 = rb[(a * 4 + 2) * HW], r3 = rb[(a * 4 + 3) * HW];
    float w = ax2 - ax1 + 1.f, h = ay2 - ay1 + 1.f;
    float cx = ax1 + 0.5f * w, cy = ay1 + 0.5f * h;
    float dx = r0 / 10.0f, dy = r1 / 10.0f;
    float dw = fminf(r2 / 5.0f, XCLIP), dh = fminf(r3 / 5.0f, XCLIP);
    float pcx = dx * w + cx, pcy = dy * h + cy;
    float pw = expf(dw) * w, ph = expf(dh) * h;
    float x1 = pcx - 0.5f * pw, y1 = pcy - 0.5f * ph;
    float x2 = pcx + 0.5f * pw - 1.f, y2 = pcy + 0.5f * ph - 1.f;
    x1 = fminf(fmaxf(x1, 0.f), IMW_M1); x2 = fminf(fmaxf(x2, 0.f), IMW_M1);
    y1 = fminf(fmaxf(y1, 0.f), IMH_M1); y2 = fminf(fmaxf(y2, 0.f), IMH_M1);
    float sc = 1.f / (1.f + expf(-logit));
    bool v = (sc > 0.f) && (x2 - x1 + 1.f >= 0.f) && (y2 - y1 + 1.f >= 0.f);
    boxes[0] = x1; boxes[1] = y1; boxes[2] = x2; boxes[3] = y2;
    float of = (float)label * CLS_OFF;
    boff[0] = x1 + of; boff[1] = y1 + of; boff[2] = x2 + of; boff[3] = y2 + of;
    scores[s] = sc; labels[s] = (unsigned)label; valid[s] = v ? 1u : 0u;
  } else {
    boxes[0] = boxes[1] = boxes[2] = boxes[3] = 0.f;
    boff[0] = boff[1] = boff[2] = boff[3] = 0.f;
    scores[s] = -1.0f; labels[s] = 0u; valid[s] = 0u;
  }
}

// ---------------- K7: IoU suppression bitmask, WMMA area-sum tiles ----------------
// One wave per 16-row x 32-col tile. Union denominator area_i+area_j computed as
// rank-2 outer sum via V_WMMA_F32_16X16X4_F32: A=[area_i,1,0,0], B=[1;area_j;0;0].
__global__ __launch_bounds__(32) void k7_mask(unsigned* __restrict__ ws) {
  int img = blockIdx.z;
  int r0 = blockIdx.y * 16;
  int c0 = blockIdx.x * 32;
  int lane = threadIdx.x;
  __shared__ float4 rbox[16];
  __shared__ float4 cbox[32];
  __shared__ float rA[16], cA[32];
  const float4* bptr = (const float4*)((const float*)(ws + BOXOFF) + (size_t)img * KPAD * 4);
  float4 cb = bptr[c0 + lane];
  cbox[lane] = cb;
  cA[lane] = (cb.z - cb.x + 1.f) * (cb.w - cb.y + 1.f);
  if (lane < 16) {
    float4 rb4 = bptr[r0 + lane];
    rbox[lane] = rb4;
    rA[lane] = (rb4.z - rb4.x + 1.f) * (rb4.w - rb4.y + 1.f);
  }
  __syncthreads();
  int n_ = lane & 15, hi = lane >> 4;
  unsigned rw[16];
#pragma unroll
  for (int r = 0; r < 16; ++r) rw[r] = 0u;
#pragma unroll
  for (int h = 0; h < 2; ++h) {
    // A 16x4: lanes 0-15 hold K0,K1 ; lanes 16-31 hold K2,K3 (set to 0)
    v2f a; a[0] = hi ? 0.f : rA[n_]; a[1] = hi ? 0.f : 1.0f;
    // B 4x16: VGPR0 = rows {0,2}, VGPR1 = rows {1,3}
    v2f b; b[0] = hi ? 0.f : 1.0f;  b[1] = hi ? 0.f : cA[h * 16 + n_];
    v8f cz = {};
    v8f S = __builtin_amdgcn_wmma_f32_16x16x4_f32(false, a, false, b, (short)0, cz,
                                                  false, false);
    float4 cc = cbox[h * 16 + n_];
#pragma unroll
    for (int v = 0; v < 8; ++v) {
      int M = v + 8 * hi; // D layout: lane L, VGPR v -> (M = v + 8*(L>=16), N = L&15)
      float4 rr = rbox[M];
      float iw = fminf(rr.z, cc.z) - fmaxf(rr.x, cc.x) + 1.f; iw = fmaxf(iw, 0.f);
      float ih = fminf(rr.w, cc.w) - fmaxf(rr.y, cc.y) + 1.f; ih = fmaxf(ih, 0.f);
      float inter = iw * ih;
      float uni = S[v] - inter;       // area_i + area_j - inter
      bool pred = (inter / uni) > 0.5f;
      unsigned bal = __builtin_amdgcn_ballot_w32(pred);
      rw[v]     |= (bal & 0xFFFFu) << (16 * h);
      rw[v + 8] |= (bal >> 16)     << (16 * h);
    }
  }
  if (lane == 0) {
    unsigned* mrow = ws + MASKW + ((size_t)img * KPAD + r0) * 32 + (c0 >> 5);
#pragma unroll
    for (int r = 0; r < 16; ++r) mrow[(size_t)r * 32] = rw[r];
  }
}

// ---------------- K8: serial greedy NMS (mask staged in 128KB LDS) + top-100 ----------------
__global__ __launch_bounds__(1024) void k8_final(unsigned* __restrict__ ws,
                                                 float* __restrict__ out) {
  extern __shared__ unsigned smem[];
  unsigned* maskLds = smem;                        // 32768 words
  float*    scoreLds = (float*)(smem + 32768);     // 1024
  unsigned* validLds = smem + 32768 + 1024;        // 1024
  float*    kK = (float*)(smem + 32768 + 2048);    // 1024
  int*      kI = (int*)(smem + 32768 + 3072);      // 1024
  int img = blockIdx.x, tid = threadIdx.x;
  const unsigned* maskG = ws + MASKW + (size_t)img * KPAD * 32;
  for (int w = tid; w < KPAD * 32; w += 1024) maskLds[w] = maskG[w];
  scoreLds[tid] = ((const float*)(ws + SCORES))[(size_t)img * KPAD + tid];
  validLds[tid] = ws[VALIDF + (size_t)img * KPAD + tid];
  kK[tid] = -1.0f; kI[tid] = tid;
  __syncthreads();
  if (tid < 32) {
    unsigned removed = 0;
    for (int i = 0; i < KTOP; ++i) {
      int w = i >> 5, b = i & 31;
      unsigned sw = (unsigned)__shfl((int)removed, w, 32);
      bool supp = (sw >> b) & 1u;
      bool ki = (validLds[i] != 0u) && !supp;
      unsigned m = maskLds[i * 32 + tid];
      if (ki) removed |= m;
      if (tid == 0 && ki) kK[i] = scoreLds[i];
    }
  }
  __syncthreads();
  bitonicSortDesc1024(kK, kI, tid);
  if (tid < 100) {
    float sc = kK[tid]; int slot = kI[tid];
    const float* bs = (const float*)(ws + BOXES) + ((size_t)img * KPAD + slot) * 4;
    float* ob = out + ((size_t)img * 100 + tid) * 4;
    ob[0] = bs[0]; ob[1] = bs[1]; ob[2] = bs[2]; ob[3] = bs[3];
    out[NIMG * 100 * 4 + img * 100 + tid] = sc;
    out[NIMG * 100 * 5 + img * 100 + tid] =
        (sc > 0.f) ? (float)ws[LABELS + (size_t)img * KPAD + slot] : 0.f;
  }
}

extern "C" void kernel_launch(void* const* d_in, const int* in_sizes, int n_in,
                              void* d_out, int out_size, void* d_ws, size_t ws_size,
                              hipStream_t stream) {
  const float* box_cls = (const float*)d_in[0];
  const float* box_reg = (const float*)d_in[1];
  const float* anchors = (const float*)d_in[2];
  unsigned* ws = (unsigned*)d_ws;   // requires ~1.4 MB of scratch
  float* out = (float*)d_out;      // [boxes 800][scores 200][labels 200]
  (void)in_sizes; (void)n_in; (void)out_size; (void)ws_size;

  k0_zero<<<512, 256, 0, stream>>>(ws);
  k1_hist<<<2048, 256, 0, stream>>>((const float4*)box_cls, ws);
  k2_sel1<<<NIMG, 1024, 0, stream>>>(ws);
  k3_hist2<<<2048, 256, 0, stream>>>((const float4*)box_cls, ws);
  k4_sel2<<<NIMG, 1024, 0, stream>>>(ws);
  k5_compact<<<2048, 256, 0, stream>>>((const float4*)box_cls, ws);
  k6a_sort<<<NIMG, 1024, 0, stream>>>(ws);
  k6_decode<<<NIMG, 1024, 0, stream>>>(box_reg, anchors, ws);
  dim3 mg(32, 64, NIMG); // col-words x row-tiles x images
  k7_mask<<<mg, 32, 0, stream>>>(ws);
  int shbytes = (32768 + 4096) * 4; // 147456 B of LDS (fits CDNA5's 320KB WGP LDS)
  hipFuncSetAttribute(reinterpret_cast<const void*>(k8_final),
                      hipFuncAttributeMaxDynamicSharedMemorySize, shbytes);
  k8_final<<<NIMG, 1024, shbytes, stream>>>(ws, out);
}